// SparseMaxPoolTestTorch_50483045597967
// MI455X (gfx1250) — compile-verified
//
#include <hip/hip_runtime.h>
#include <hip/hip_bf16.h>
#include <stdint.h>

// Composed geometry: two (k=3,s=2,p=1) max-pools over 96^3 == one max over
// x in [4*o-3, 4*o+3] clipped to [0,95], output 24^3 per batch, 64 channels.
#define IN_D   96
#define OUT_D  24
#define NCH    64

// ---------------------------------------------------------------------------
// Kernel 1: fill output with -inf (vectorized b128 stores).
// ---------------------------------------------------------------------------
__global__ void init_neg_inf_kernel(float4* __restrict__ out, int n4) {
    int i = blockIdx.x * blockDim.x + threadIdx.x;
    if (i < n4) {
        const float ninf = -__builtin_inff();
        out[i] = make_float4(ninf, ninf, ninf, ninf);
    }
}

// ---------------------------------------------------------------------------
// CDNA5 float-max atomic, device scope (RMW must be coherent across WGPs).
// ---------------------------------------------------------------------------
__device__ __forceinline__ void atomic_max_f32_dev(float* addr, float val) {
    asm volatile("global_atomic_max_num_f32 %0, %1, off scope:SCOPE_DEV"
                 :: "v"(addr), "v"(val)
                 : "memory");
}

// ---------------------------------------------------------------------------
// Kernel 2: direct scatter-max of each (point, channel) into the <=8 output
// cells whose composed 7^3 window covers the point. One thread per
// (point, channel); each wave32 covers 32 channels of a single point, so the
// point index is wave-uniform -> scalar loads for the coordinates.
// ---------------------------------------------------------------------------
__global__ void scatter_max_kernel(const float* __restrict__ feat,
                                   const int*   __restrict__ coors,
                                   float*       __restrict__ out,
                                   int npts) {
    int t = blockIdx.x * blockDim.x + threadIdx.x;
    int p = t >> 6;          // point index   (uniform across the wave)
    int c = t & (NCH - 1);   // channel index (per lane)
    if (p >= npts) return;
    p = __builtin_amdgcn_readfirstlane(p);   // force SGPR -> s_load coords

    const int b = coors[p * 4 + 0];
    const int x = coors[p * 4 + 1];
    const int y = coors[p * 4 + 2];
    const int z = coors[p * 4 + 3];

    const float v = feat[p * NCH + c];

    // per-dim output range: ceil((x-3)/4) = x>>2 (x>=0), floor((x+3)/4) clamped
    const int x0 = x >> 2, x1 = min((x + 3) >> 2, OUT_D - 1);
    const int y0 = y >> 2, y1 = min((y + 3) >> 2, OUT_D - 1);
    const int z0 = z >> 2, z1 = min((z + 3) >> 2, OUT_D - 1);

    float* base = out + (size_t)b * (OUT_D * OUT_D * OUT_D * NCH) + c;
    for (int ox = x0; ox <= x1; ++ox) {
        for (int oy = y0; oy <= y1; ++oy) {
            for (int oz = z0; oz <= z1; ++oz) {
                atomic_max_f32_dev(base + ((ox * OUT_D + oy) * OUT_D + oz) * NCH, v);
            }
        }
    }
}

// ---------------------------------------------------------------------------
// Kernel 3: -inf -> 0 rewrite. Stages each block's tile through LDS with the
// gfx1250 async-to-LDS path (global_load_async_to_lds_b128 + s_wait_asynccnt),
// then transforms and stores b128.
// ---------------------------------------------------------------------------
__global__ void finalize_kernel(float* __restrict__ out, int n4) {
    __shared__ float4 tile[256];
    int i = blockIdx.x * blockDim.x + threadIdx.x;
    if (i >= n4) return;

    const float4* src = ((const float4*)out) + i;
    // Low 32 bits of a flat shared pointer == LDS byte offset (ISA aperture map).
    uint32_t ldsoff = (uint32_t)(uintptr_t)(&tile[threadIdx.x]);

    asm volatile("global_load_async_to_lds_b128 %0, %1, off"
                 :: "v"(ldsoff), "v"(src)
                 : "memory");
    asm volatile("s_wait_asynccnt 0" ::: "memory");  // own-wave data -> no barrier

    float4 v = tile[threadIdx.x];
    const float ninf = -__builtin_inff();
    v.x = (v.x == ninf) ? 0.0f : v.x;
    v.y = (v.y == ninf) ? 0.0f : v.y;
    v.z = (v.z == ninf) ? 0.0f : v.z;
    v.w = (v.w == ninf) ? 0.0f : v.w;
    ((float4*)out)[i] = v;
}

// ---------------------------------------------------------------------------
extern "C" void kernel_launch(void* const* d_in, const int* in_sizes, int n_in,
                              void* d_out, int out_size, void* d_ws, size_t ws_size,
                              hipStream_t stream) {
    const float* feat  = (const float*)d_in[0];   // (NPTS, 64) f32
    const int*   coors = (const int*)  d_in[1];   // (NPTS, 4)  i32 [b,x,y,z]
    float*       out   = (float*)d_out;           // (B,24,24,24,64) f32

    const int npts = in_sizes[0] / NCH;
    const int n4   = out_size / 4;                // float4 count (exactly divisible)

    const int blk = 256;
    const int initBlocks = (n4 + blk - 1) / blk;

    // 1) output <- -inf (must run every call: harness poisons/reuses d_out)
    init_neg_inf_kernel<<<initBlocks, blk, 0, stream>>>((float4*)out, n4);

    // 2) fused scatter + (pool o pool) via device-scope float-max atomics
    const int total = npts * NCH;
    scatter_max_kernel<<<(total + blk - 1) / blk, blk, 0, stream>>>(feat, coors, out, npts);

    // 3) -inf -> 0 (async-to-LDS staged)
    finalize_kernel<<<initBlocks, blk, 0, stream>>>(out, n4);
}